// SPCC_64518998721095
// MI455X (gfx1250) — compile-verified
//
#include <hip/hip_runtime.h>
#include <math.h>

#define CCH 128
#define NEG_SLOPE 0.2f

typedef float v2f __attribute__((ext_vector_type(2)));
typedef float v8f __attribute__((ext_vector_type(8)));

__device__ __forceinline__ float lrelu(float x) {
    return x > 0.f ? x : NEG_SLOPE * x;
}

// Monotonic float <-> uint mapping so atomicMax(uint) == float max.
// Encoding of every real float is > 0, so a zero-filled buffer acts as -inf.
__device__ __forceinline__ unsigned fenc(float f) {
    unsigned u = __float_as_uint(f);
    return u ^ ((u & 0x80000000u) ? 0xFFFFFFFFu : 0x80000000u);
}
__device__ __forceinline__ float fdec(unsigned u) {
    unsigned b = (u & 0x80000000u) ? (u ^ 0x80000000u) : ~u;
    return __uint_as_float(b);
}

// ---------------------------------------------------------------------------
// Zero-fill (graph-capture-safe init; harness poisons d_out / d_ws with 0xAA)
// ---------------------------------------------------------------------------
__global__ void zero_kernel(float* __restrict__ p, size_t n) {
    size_t i = (size_t)blockIdx.x * blockDim.x + threadIdx.x;
    size_t stride = (size_t)gridDim.x * blockDim.x;
    for (; i < n; i += stride) p[i] = 0.f;
}

// ---------------------------------------------------------------------------
// Y[N,128] = X[N,128] @ W[128,128], fp32 WMMA 16x16x4.
// W is staged transposed into LDS so each B fragment is one 8-byte ds load.
// 8 waves per block; each wave computes a 16-row x 128-col strip:
// 8 accumulator tiles x 32 K-steps = 256 v_wmma per wave.
// Requires N % 16 == 0 (holds: 100000, 20000).
// ---------------------------------------------------------------------------
__global__ __launch_bounds__(256)
void gemm128_wmma(const float* __restrict__ X, const float* __restrict__ W,
                  float* __restrict__ Y, int N) {
    __shared__ float WT[CCH * CCH];       // WT[n*128 + k] = W[k*128 + n]
    for (int i = threadIdx.x; i < CCH * CCH; i += 256) {
        int k = i >> 7;
        int n = i & 127;
        WT[n * CCH + k] = W[i];
    }
    __syncthreads();

    const int wave = threadIdx.x >> 5;
    const int lane = threadIdx.x & 31;
    const int row0 = (blockIdx.x * 8 + wave) * 16;
    if (row0 >= N) return;

    const int mrow = lane & 15;           // A: M = mrow ; B/C/D: N-in-tile = mrow
    const int koff = (lane >> 4) << 1;    // 0 or 2 (K pair held by this half-wave)
    const float* __restrict__ Xrow = X + (size_t)(row0 + mrow) * CCH;

    v8f acc[8];
    #pragma unroll
    for (int t = 0; t < 8; ++t)
        #pragma unroll
        for (int j = 0; j < 8; ++j) acc[t][j] = 0.f;

    for (int k0 = 0; k0 < CCH; k0 += 4) {
        // A fragment: lane holds X[row0+mrow][k0+koff .. +1]  (8B, aligned)
        v2f a = *(const v2f*)(Xrow + k0 + koff);
        #pragma unroll
        for (int t = 0; t < 8; ++t) {
            const int n = t * 16 + mrow;
            // B fragment: lane holds W[k0+koff .. +1][n] == WT[n][k0+koff .. +1]
            v2f b = *(const v2f*)(&WT[n * CCH + k0 + koff]);
            acc[t] = __builtin_amdgcn_wmma_f32_16x16x4_f32(
                false, a, false, b, (short)0, acc[t], false, false);
        }
    }

    // C/D layout: VGPR v -> row (v) for lanes 0-15, row (v+8) for lanes 16-31
    const int rbase = row0 + ((lane >> 4) << 3);
    #pragma unroll
    for (int t = 0; t < 8; ++t) {
        const int col = t * 16 + mrow;
        #pragma unroll
        for (int v = 0; v < 8; ++v)
            Y[(size_t)(rbase + v) * CCH + col] = acc[t][v];
    }
}

// ---------------------------------------------------------------------------
// Two dot products per 128-float row: s0[r] = Y[r]·a0 ; s1[r] = Y[r]·a1.
// One wave per row, float4 per lane, xor-shuffle tree reduction (wave32).
// ---------------------------------------------------------------------------
__global__ __launch_bounds__(256)
void rowdot2(const float* __restrict__ Y, const float* __restrict__ a0,
             const float* __restrict__ a1, float* __restrict__ s0,
             float* __restrict__ s1, int N) {
    const int wave = threadIdx.x >> 5;
    const int lane = threadIdx.x & 31;
    const int r = blockIdx.x * 8 + wave;
    if (r >= N) return;
    const float4 v  = ((const float4*)(Y + (size_t)r * CCH))[lane];
    const float4 w0 = ((const float4*)a0)[lane];
    const float4 w1 = ((const float4*)a1)[lane];
    float d0 = v.x * w0.x + v.y * w0.y + v.z * w0.z + v.w * w0.w;
    float d1 = v.x * w1.x + v.y * w1.y + v.z * w1.z + v.w * w1.w;
    #pragma unroll
    for (int off = 16; off > 0; off >>= 1) {
        d0 += __shfl_xor(d0, off);
        d1 += __shfl_xor(d1, off);
    }
    if (lane == 0) { s0[r] = d0; s1[r] = d1; }
}

// ---------------------------------------------------------------------------
// Segment softmax over edges, score e = lrelu(vA[iA[e]] + vB[iB[e]]).
// Pass 1: per-segment max (atomicMax on monotonic uint encoding).
// Pass 2: per-segment sum of exp(e - max).
// Pass 3: wave-per-edge weighted scatter of a 128-float source row.
// ---------------------------------------------------------------------------
__global__ void seg_max_kernel(const float* __restrict__ vA, const int* __restrict__ iA,
                               const float* __restrict__ vB, const int* __restrict__ iB,
                               const int* __restrict__ seg, unsigned* __restrict__ maxb,
                               int E) {
    int i = blockIdx.x * blockDim.x + threadIdx.x;
    int stride = gridDim.x * blockDim.x;
    for (; i < E; i += stride) {
        float e = lrelu(vA[iA[i]] + vB[iB[i]]);
        atomicMax(&maxb[seg[i]], fenc(e));
    }
}

__global__ void seg_denom_kernel(const float* __restrict__ vA, const int* __restrict__ iA,
                                 const float* __restrict__ vB, const int* __restrict__ iB,
                                 const int* __restrict__ seg, const unsigned* __restrict__ maxb,
                                 float* __restrict__ denom, int E) {
    int i = blockIdx.x * blockDim.x + threadIdx.x;
    int stride = gridDim.x * blockDim.x;
    for (; i < E; i += stride) {
        float e = lrelu(vA[iA[i]] + vB[iB[i]]);
        int s = seg[i];
        atomicAdd(&denom[s], __expf(e - fdec(maxb[s])));
    }
}

__global__ __launch_bounds__(256)
void seg_aggregate_kernel(const float* __restrict__ vA, const int* __restrict__ iA,
                          const float* __restrict__ vB, const int* __restrict__ iB,
                          const int* __restrict__ seg, const unsigned* __restrict__ maxb,
                          const float* __restrict__ denom,
                          const float* __restrict__ Msrc, const int* __restrict__ srcIdx,
                          float* __restrict__ Out, int E) {
    const int lane = threadIdx.x & 31;
    const int w = blockIdx.x * 8 + (threadIdx.x >> 5);   // one wave per edge
    if (w >= E) return;
    const int s = seg[w];
    const float e = lrelu(vA[iA[w]] + vB[iB[w]]);
    const float att = __expf(e - fdec(maxb[s])) / denom[s];
    const float4 v = ((const float4*)(Msrc + (size_t)srcIdx[w] * CCH))[lane];
    float* dst = Out + (size_t)s * CCH + lane * 4;
    atomicAdd(dst + 0, att * v.x);
    atomicAdd(dst + 1, att * v.y);
    atomicAdd(dst + 2, att * v.z);
    atomicAdd(dst + 3, att * v.w);
}

// out0 = relu(acc_in_out0) + relu(accB)   (outer relu of a sum of relus == sum)
__global__ void final_sum_relu(float* __restrict__ out, const float* __restrict__ accB,
                               size_t n) {
    size_t i = (size_t)blockIdx.x * blockDim.x + threadIdx.x;
    size_t stride = (size_t)gridDim.x * blockDim.x;
    for (; i < n; i += stride)
        out[i] = fmaxf(out[i], 0.f) + fmaxf(accB[i], 0.f);
}

__global__ void final_relu(float* __restrict__ out, size_t n) {
    size_t i = (size_t)blockIdx.x * blockDim.x + threadIdx.x;
    size_t stride = (size_t)gridDim.x * blockDim.x;
    for (; i < n; i += stride) out[i] = fmaxf(out[i], 0.f);
}

// ---------------------------------------------------------------------------
extern "C" void kernel_launch(void* const* d_in, const int* in_sizes, int n_in,
                              void* d_out, int out_size, void* d_ws, size_t ws_size,
                              hipStream_t stream) {
    const float* x0     = (const float*)d_in[0];
    const float* x2     = (const float*)d_in[1];
    const float* w_hbs  = (const float*)d_in[2];
    const float* a_hbs  = (const float*)d_in[3];
    const float* w_s    = (const float*)d_in[4];
    const float* w_t    = (const float*)d_in[5];
    const float* a_hbns = (const float*)d_in[6];
    const int* adj_r    = (const int*)d_in[7];
    const int* adj_c    = (const int*)d_in[8];
    const int* inc_r    = (const int*)d_in[9];
    const int* inc_c    = (const int*)d_in[10];

    const size_t n0 = (size_t)in_sizes[0] / CCH;   // 100000
    const size_t n2 = (size_t)in_sizes[1] / CCH;   // 20000
    const int e_adj = in_sizes[7];                 // 3,200,000
    const int e_inc = in_sizes[9];                 // 1,000,000

    float* out0 = (float*)d_out;                   // [n0,128] : x_0_to_0 accum
    float* out2 = out0 + n0 * CCH;                 // [n2,128] : x_0_to_2 accum

    // ---- workspace layout (floats) ----
    float* ws    = (float*)d_ws;
    float* m0    = ws;                   // n0*128
    float* tmsg  = m0   + n0 * CCH;      // n0*128
    float* smsg  = tmsg + n0 * CCH;      // n2*128
    float* accB  = smsg + n2 * CCH;      // n0*128  (x_2_to_0 accum)
    float* si    = accB + n0 * CCH;      // n0
    float* sj    = si + n0;              // n0
    float* ta_lo = sj + n0;              // n0  t_msg @ a_hbns[:C]
    float* ta_hi = ta_lo + n0;           // n0  t_msg @ a_hbns[C:]
    float* sa_lo = ta_hi + n0;           // n2  s_msg @ a_hbns[:C]
    float* sa_hi = sa_lo + n2;           // n2  s_msg @ a_hbns[C:]
    unsigned* max_adj = (unsigned*)(sa_hi + n2);   // n0
    float*    den_adj = (float*)(max_adj + n0);    // n0
    unsigned* max_st  = (unsigned*)(den_adj + n0); // n0
    float*    den_st  = (float*)(max_st + n0);     // n0
    unsigned* max_ts  = (unsigned*)(den_st + n0);  // n2
    float*    den_ts  = (float*)(max_ts + n2);     // n2
    (void)den_ts; (void)ws_size; (void)n_in;

    // Zero the accumulators / softmax state (encoded-max sentinel 0 == -inf).
    const size_t zeroN = n0 * CCH + 8 * n0 + 4 * n2;   // accB .. den_ts
    zero_kernel<<<2048, 256, 0, stream>>>(accB, zeroN);
    zero_kernel<<<2048, 256, 0, stream>>>((float*)d_out, (size_t)out_size);

    // ---- dense GEMMs on WMMA pipes (fp32) ----
    const int gblk0 = (int)((n0 / 16 + 7) / 8);
    const int gblk2 = (int)((n2 / 16 + 7) / 8);
    gemm128_wmma<<<gblk0, 256, 0, stream>>>(x0, w_hbs, m0,   (int)n0);
    gemm128_wmma<<<gblk0, 256, 0, stream>>>(x0, w_t,   tmsg, (int)n0);
    gemm128_wmma<<<gblk2, 256, 0, stream>>>(x2, w_s,   smsg, (int)n2);

    // ---- per-row attention scalars ----
    const int rblk0 = (int)((n0 + 7) / 8);
    const int rblk2 = (int)((n2 + 7) / 8);
    rowdot2<<<rblk0, 256, 0, stream>>>(m0,   a_hbs,  a_hbs  + CCH, si,    sj,    (int)n0);
    rowdot2<<<rblk0, 256, 0, stream>>>(tmsg, a_hbns, a_hbns + CCH, ta_lo, ta_hi, (int)n0);
    rowdot2<<<rblk2, 256, 0, stream>>>(smsg, a_hbns, a_hbns + CCH, sa_lo, sa_hi, (int)n2);

    const int sblkA = (e_adj + 255) / 256;
    const int sblkI = (e_inc + 255) / 256;
    const int ablkA = (e_adj + 7) / 8;
    const int ablkI = (e_inc + 7) / 8;

    // ---- HBS: 0-cells -> 0-cells, e = lrelu(si[r] + sj[c]), segments = rows ----
    seg_max_kernel  <<<sblkA, 256, 0, stream>>>(si, adj_r, sj, adj_c, adj_r, max_adj, e_adj);
    seg_denom_kernel<<<sblkA, 256, 0, stream>>>(si, adj_r, sj, adj_c, adj_r, max_adj, den_adj, e_adj);
    seg_aggregate_kernel<<<ablkA, 256, 0, stream>>>(si, adj_r, sj, adj_c, adj_r,
                                                    max_adj, den_adj, m0, adj_c, out0, e_adj);

    // ---- HBNS e_st: e = lrelu(sa_lo[c] + ta_hi[r]), segments = inc_rows (0-cells) ----
    seg_max_kernel  <<<sblkI, 256, 0, stream>>>(sa_lo, inc_c, ta_hi, inc_r, inc_r, max_st, e_inc);
    seg_denom_kernel<<<sblkI, 256, 0, stream>>>(sa_lo, inc_c, ta_hi, inc_r, inc_r, max_st, den_st, e_inc);
    seg_aggregate_kernel<<<ablkI, 256, 0, stream>>>(sa_lo, inc_c, ta_hi, inc_r, inc_r,
                                                    max_st, den_st, smsg, inc_c, accB, e_inc);

    // ---- HBNS e_ts: e = lrelu(ta_lo[r] + sa_hi[c]), segments = inc_cols (2-cells) ----
    seg_max_kernel  <<<sblkI, 256, 0, stream>>>(ta_lo, inc_r, sa_hi, inc_c, inc_c, max_ts, e_inc);
    seg_denom_kernel<<<sblkI, 256, 0, stream>>>(ta_lo, inc_r, sa_hi, inc_c, inc_c, max_ts, den_ts, e_inc);
    seg_aggregate_kernel<<<ablkI, 256, 0, stream>>>(ta_lo, inc_r, sa_hi, inc_c, inc_c,
                                                    max_ts, den_ts, tmsg, inc_r, out2, e_inc);

    // ---- combine ----
    final_sum_relu<<<2048, 256, 0, stream>>>(out0, accB, n0 * CCH);
    final_relu    <<<2048, 256, 0, stream>>>(out2, n2 * CCH);
}